// MultiRelationGNN_61143154426125
// MI455X (gfx1250) — compile-verified
//
#include <hip/hip_runtime.h>
#include <hip/hip_bf16.h>

typedef __attribute__((ext_vector_type(2))) float v2f;
typedef __attribute__((ext_vector_type(8))) float v8f;

#define WMMA_F32X4(A, B, C) \
  __builtin_amdgcn_wmma_f32_16x16x4_f32(false, (A), false, (B), (short)0, (C), false, false)

#define HDIM 64
#define RREL 5
#define TBD 12

// ---------------------------------------------------------------------------
// e0 = x @ field_W + field_b     ( [N,128] @ [128,64] )
// one wave per 16-row tile; 4 column tiles of 16; K = 128 in steps of 4
// ---------------------------------------------------------------------------
__global__ __launch_bounds__(256) void k_field(const float* __restrict__ x,
                                               const float* __restrict__ W,
                                               const float* __restrict__ bias,
                                               float* __restrict__ e0, int n_tiles) {
  int wave = threadIdx.x >> 5;
  int lane = threadIdx.x & 31;
  int tile = blockIdx.x * (blockDim.x >> 5) + wave;
  if (tile >= n_tiles) return;            // wave-uniform: EXEC stays all-1 for WMMA
  int m0 = tile * 16;
  int hi = lane >> 4;                      // which K half-pair this lane holds
  int ml = lane & 15;                      // A row / B,C column within tile

  v8f acc[4];
#pragma unroll
  for (int nt = 0; nt < 4; ++nt)
#pragma unroll
    for (int i = 0; i < 8; ++i) acc[nt][i] = 0.0f;

  const float* arow = x + (size_t)(m0 + ml) * 128;
  for (int kk = 0; kk < 128; kk += 4) {
    int ka = kk + 2 * hi;
    v2f a; a.x = arow[ka]; a.y = arow[ka + 1];
#pragma unroll
    for (int nt = 0; nt < 4; ++nt) {
      int n = nt * 16 + ml;
      v2f b; b.x = W[(size_t)ka * HDIM + n]; b.y = W[(size_t)(ka + 1) * HDIM + n];
      acc[nt] = WMMA_F32X4(a, b, acc[nt]);
    }
  }
#pragma unroll
  for (int nt = 0; nt < 4; ++nt) {
    int n = nt * 16 + ml;
    float bv = bias[n];
#pragma unroll
    for (int v = 0; v < 8; ++v) {
      int row = m0 + v + 8 * hi;
      e0[(size_t)row * HDIM + n] = acc[nt][v] + bv;
    }
  }
}

// ---------------------------------------------------------------------------
// per-edge time-decay weight: w[e] = lambda * exp(-(edge_time[e,:] . beta))
// Stage 256 edges (12 KB) through LDS so the global reads are fully coalesced.
// ---------------------------------------------------------------------------
__global__ __launch_bounds__(256) void k_edge_w(const float* __restrict__ edge_time,
                                                const float* __restrict__ beta,
                                                const float* __restrict__ lambda_p,
                                                float* __restrict__ w, int E) {
  __shared__ float et[256 * TBD];
  int e0 = blockIdx.x * 256;                 // first edge of this block
  int tid = threadIdx.x;
  // coalesced staging: 3072 contiguous floats
  int avail = E - e0;                        // edges this block actually covers
  int nflt = (avail >= 256 ? 256 : avail) * TBD;
  size_t base = (size_t)e0 * TBD;
#pragma unroll
  for (int i = 0; i < TBD; ++i) {
    int idx = tid + i * 256;
    if (idx < nflt) et[idx] = edge_time[base + idx];
  }
  __syncthreads();
  int e = e0 + tid;
  if (e >= E) return;
  float logit = 0.0f;
#pragma unroll
  for (int j = 0; j < TBD; ++j) logit += et[tid * TBD + j] * beta[j];
  w[e] = lambda_p[0] * expf(-logit);
}

// ---------------------------------------------------------------------------
// zero scratch (s then t, contiguous)
// ---------------------------------------------------------------------------
__global__ __launch_bounds__(256) void k_zero(float* __restrict__ p, size_t n) {
  size_t i = (size_t)blockIdx.x * blockDim.x + threadIdx.x;
  if (i < n) p[i] = 0.0f;
}

// ---------------------------------------------------------------------------
// scatter: t[dst, r, h] += w_e * ein[src, h];  s[dst, r] += w_e
// One wave per edge; edge metadata made wave-uniform via readfirstlane so it
// lowers to scalar (SMEM) loads; each lane handles features lane and lane+32.
// ---------------------------------------------------------------------------
__global__ __launch_bounds__(256) void k_scatter(const float* __restrict__ ein,
                                                 const int* __restrict__ edge_index,
                                                 const int* __restrict__ edge_type,
                                                 const float* __restrict__ w,
                                                 float* __restrict__ t,
                                                 float* __restrict__ s, int E) {
  int wid = (int)(((long long)blockIdx.x * blockDim.x + threadIdx.x) >> 5);
  int lane = threadIdx.x & 31;
  if (wid >= E) return;
  int e = __builtin_amdgcn_readfirstlane(wid);   // wave-uniform edge id
  int srcn = edge_index[e];                      // scalar loads
  int dstn = edge_index[(size_t)E + e];
  int r = edge_type[e];
  float we = w[e];
  const float* src_row = ein + (size_t)srcn * HDIM;
  float* trow = t + ((size_t)dstn * RREL + r) * HDIM;
  atomicAdd(&trow[lane], we * src_row[lane]);
  atomicAdd(&trow[lane + 32], we * src_row[lane + 32]);
  if (lane == 0) atomicAdd(&s[(size_t)dstn * RREL + r], we);
}

// ---------------------------------------------------------------------------
// combine:  eout = sum_r t[:,r,:] @ W1_r + (diag(s_r) * ein) @ W2_r + s_r (x) b_r
// relW[r] is [128,64] with rows 0..63 = W1 (x_j side), rows 64..127 = W2 (x_i side)
// ---------------------------------------------------------------------------
__global__ __launch_bounds__(256) void k_combine(const float* __restrict__ ein,
                                                 const float* __restrict__ t,
                                                 const float* __restrict__ s,
                                                 const float* __restrict__ relW,
                                                 const float* __restrict__ relb,
                                                 float* __restrict__ eout, int n_tiles) {
  int wave = threadIdx.x >> 5;
  int lane = threadIdx.x & 31;
  int tile = blockIdx.x * (blockDim.x >> 5) + wave;
  if (tile >= n_tiles) return;
  int m0 = tile * 16;
  int hi = lane >> 4;
  int ml = lane & 15;

  v8f acc[4];
#pragma unroll
  for (int nt = 0; nt < 4; ++nt)
#pragma unroll
    for (int i = 0; i < 8; ++i) acc[nt][i] = 0.0f;

  const float* earow = ein + (size_t)(m0 + ml) * HDIM;
  for (int r = 0; r < RREL; ++r) {
    const float* trow = t + ((size_t)(m0 + ml) * RREL + r) * HDIM;
    float srow = s[(size_t)(m0 + ml) * RREL + r];
    const float* W1 = relW + (size_t)r * 128 * HDIM;
    const float* W2 = W1 + (size_t)64 * HDIM;
    for (int kk = 0; kk < HDIM; kk += 4) {
      int ka = kk + 2 * hi;
      v2f a1; a1.x = trow[ka];         a1.y = trow[ka + 1];
      v2f a2; a2.x = earow[ka] * srow; a2.y = earow[ka + 1] * srow;
#pragma unroll
      for (int nt = 0; nt < 4; ++nt) {
        int n = nt * 16 + ml;
        v2f b1; b1.x = W1[(size_t)ka * HDIM + n]; b1.y = W1[(size_t)(ka + 1) * HDIM + n];
        acc[nt] = WMMA_F32X4(a1, b1, acc[nt]);
        v2f b2; b2.x = W2[(size_t)ka * HDIM + n]; b2.y = W2[(size_t)(ka + 1) * HDIM + n];
        acc[nt] = WMMA_F32X4(a2, b2, acc[nt]);
      }
    }
  }
#pragma unroll
  for (int nt = 0; nt < 4; ++nt) {
    int n = nt * 16 + ml;
#pragma unroll
    for (int v = 0; v < 8; ++v) {
      int row = m0 + v + 8 * hi;
      float bias = 0.0f;
#pragma unroll
      for (int r = 0; r < RREL; ++r)
        bias += s[(size_t)row * RREL + r] * relb[(size_t)r * HDIM + n];
      eout[(size_t)row * HDIM + n] = acc[nt][v] + bias;
    }
  }
}

// ---------------------------------------------------------------------------
// out = lrelu(e2@W2+b2) + lrelu(e1@W1+b1) + lrelu(e0@W0+b0)
// ---------------------------------------------------------------------------
__global__ __launch_bounds__(256) void k_out(const float* __restrict__ e0,
                                             const float* __restrict__ e1,
                                             const float* __restrict__ e2,
                                             const float* __restrict__ W0, const float* __restrict__ b0,
                                             const float* __restrict__ W1, const float* __restrict__ b1,
                                             const float* __restrict__ W2, const float* __restrict__ b2,
                                             float* __restrict__ out, int n_tiles) {
  int wave = threadIdx.x >> 5;
  int lane = threadIdx.x & 31;
  int tile = blockIdx.x * (blockDim.x >> 5) + wave;
  if (tile >= n_tiles) return;
  int m0 = tile * 16;
  int hi = lane >> 4;
  int ml = lane & 15;

  const float* a0row = e0 + (size_t)(m0 + ml) * HDIM;
  const float* a1row = e1 + (size_t)(m0 + ml) * HDIM;
  const float* a2row = e2 + (size_t)(m0 + ml) * HDIM;

  for (int nt = 0; nt < 4; ++nt) {
    int n = nt * 16 + ml;
    v8f c0, c1, c2;
#pragma unroll
    for (int i = 0; i < 8; ++i) { c0[i] = 0.0f; c1[i] = 0.0f; c2[i] = 0.0f; }
    for (int kk = 0; kk < HDIM; kk += 4) {
      int ka = kk + 2 * hi;
      v2f a0; a0.x = a0row[ka]; a0.y = a0row[ka + 1];
      v2f a1; a1.x = a1row[ka]; a1.y = a1row[ka + 1];
      v2f a2; a2.x = a2row[ka]; a2.y = a2row[ka + 1];
      v2f f0; f0.x = W0[(size_t)ka * HDIM + n]; f0.y = W0[(size_t)(ka + 1) * HDIM + n];
      v2f f1; f1.x = W1[(size_t)ka * HDIM + n]; f1.y = W1[(size_t)(ka + 1) * HDIM + n];
      v2f f2; f2.x = W2[(size_t)ka * HDIM + n]; f2.y = W2[(size_t)(ka + 1) * HDIM + n];
      c0 = WMMA_F32X4(a0, f0, c0);
      c1 = WMMA_F32X4(a1, f1, c1);
      c2 = WMMA_F32X4(a2, f2, c2);
    }
    float bb0 = b0[n], bb1 = b1[n], bb2 = b2[n];
#pragma unroll
    for (int v = 0; v < 8; ++v) {
      int row = m0 + v + 8 * hi;
      float y0 = c0[v] + bb0; y0 = (y0 > 0.0f) ? y0 : 0.01f * y0;
      float y1 = c1[v] + bb1; y1 = (y1 > 0.0f) ? y1 : 0.01f * y1;
      float y2 = c2[v] + bb2; y2 = (y2 > 0.0f) ? y2 : 0.01f * y2;
      out[(size_t)row * HDIM + n] = y0 + y1 + y2;
    }
  }
}

// ---------------------------------------------------------------------------
extern "C" void kernel_launch(void* const* d_in, const int* in_sizes, int n_in,
                              void* d_out, int out_size, void* d_ws, size_t ws_size,
                              hipStream_t stream) {
  const float* x       = (const float*)d_in[0];
  const int*   eidx    = (const int*)d_in[1];
  const int*   etype   = (const int*)d_in[2];
  const float* etime   = (const float*)d_in[3];
  const float* lambda_ = (const float*)d_in[4];
  const float* beta    = (const float*)d_in[5];
  const float* fieldW  = (const float*)d_in[6];
  const float* fieldb  = (const float*)d_in[7];
  const float* rel1W   = (const float*)d_in[8];
  const float* rel1b   = (const float*)d_in[9];
  const float* rel2W   = (const float*)d_in[10];
  const float* rel2b   = (const float*)d_in[11];
  const float* out0W   = (const float*)d_in[12];
  const float* out0b   = (const float*)d_in[13];
  const float* out1W   = (const float*)d_in[14];
  const float* out1b   = (const float*)d_in[15];
  const float* out2W   = (const float*)d_in[16];
  const float* out2b   = (const float*)d_in[17];

  const int N = in_sizes[0] / 128;   // 50000
  const int E = in_sizes[2];         // 800000
  const int n_tiles = N / 16;        // 3125 (N divisible by 16)

  // workspace layout (fp32): w[E] | s[N*5] | t[N*5*64] | e0 | e1 | e2
  float* ws   = (float*)d_ws;
  float* wbuf = ws;
  float* sbuf = wbuf + (size_t)E;
  float* tbuf = sbuf + (size_t)N * RREL;
  float* e0   = tbuf + (size_t)N * RREL * HDIM;
  float* e1   = e0 + (size_t)N * HDIM;
  float* e2   = e1 + (size_t)N * HDIM;
  const size_t st_count = (size_t)N * RREL * (HDIM + 1);   // s + t contiguous

  dim3 blk(256);
  int tile_blocks = (n_tiles + 7) / 8;                 // 8 waves per block
  int zero_blocks = (int)((st_count + 255) / 256);
  long long sc_thr = (long long)E * 32;                // one wave per edge
  int sc_blocks = (int)((sc_thr + 255) / 256);

  // node embedding
  k_field<<<tile_blocks, blk, 0, stream>>>(x, fieldW, fieldb, e0, n_tiles);
  // edge weights (shared by both layers)
  k_edge_w<<<(E + 255) / 256, blk, 0, stream>>>(etime, beta, lambda_, wbuf, E);

  // layer 1
  k_zero<<<zero_blocks, blk, 0, stream>>>(sbuf, st_count);
  k_scatter<<<sc_blocks, blk, 0, stream>>>(e0, eidx, etype, wbuf, tbuf, sbuf, E);
  k_combine<<<tile_blocks, blk, 0, stream>>>(e0, tbuf, sbuf, rel1W, rel1b, e1, n_tiles);

  // layer 2
  k_zero<<<zero_blocks, blk, 0, stream>>>(sbuf, st_count);
  k_scatter<<<sc_blocks, blk, 0, stream>>>(e1, eidx, etype, wbuf, tbuf, sbuf, E);
  k_combine<<<tile_blocks, blk, 0, stream>>>(e1, tbuf, sbuf, rel2W, rel2b, e2, n_tiles);

  // fused output head
  k_out<<<tile_blocks, blk, 0, stream>>>(e0, e1, e2, out0W, out0b, out1W, out1b,
                                         out2W, out2b, (float*)d_out, n_tiles);
}